// Conv2D_20701742366871
// MI455X (gfx1250) — compile-verified
//
#include <hip/hip_runtime.h>
#include <hip/hip_bf16.h>

typedef __attribute__((ext_vector_type(2))) float v2f;
typedef __attribute__((ext_vector_type(8))) float v8f;

#define IN_H 4096
#define IN_W 4096
#define OUT_H 4092
#define OUT_W 4092

// Block: 256 threads = 8 wave32; block output tile 32 (rows) x 64 (cols),
// wave (r,c) in 2x4 arrangement each owning a 16x16 WMMA tile.
// LDS input tile: 36 rows x 68 cols (4-halo on each trailing edge).
#define LDS_R 36
#define LDS_C 68

__global__ __launch_bounds__(256) void conv5x5_wmma_f32(
    const float* __restrict__ x, const float* __restrict__ wgt,
    const float* __restrict__ bias, float* __restrict__ out) {
  __shared__ float tile[LDS_R][LDS_C];
  __shared__ float wl[26];

  const int tid  = (int)threadIdx.x;
  const int lane = tid & 31;
  const int wave = tid >> 5;
  const int m    = lane & 15;   // M row of A / N col of B / N col of C
  const int half = lane >> 4;   // selects K pair {0,1} vs {2,3}

  const int bx = (int)blockIdx.x * 64;  // output col origin of block
  const int by = (int)blockIdx.y * 32;  // output row origin of block

  if (tid < 25) wl[tid] = wgt[tid];
  if (tid == 25) wl[25] = bias[0];

  // Stage 36x68 input tile into LDS, coords clamped (only affects outputs
  // that are predicated off at the store).
  for (int idx = tid; idx < LDS_R * LDS_C; idx += 256) {
    int r = idx / LDS_C;
    int c = idx - r * LDS_C;
    int gy = by + r; gy = gy > (IN_H - 1) ? (IN_H - 1) : gy;
    int gx = bx + c; gx = gx > (IN_W - 1) ? (IN_W - 1) : gx;
    tile[r][c] = x[gy * IN_W + gx];
  }
  __syncthreads();

  // Per-lane banded-Toeplitz B values: B_ky[k][n] = w[ky][k-n] for 0<=k-n<5.
  // This lane supplies k = kc*4 + 2*half (+0 in VGPR0, +1 in VGPR1), n = m.
  float bx0[5][5], bx1[5][5];
#pragma unroll
  for (int ky = 0; ky < 5; ++ky) {
#pragma unroll
    for (int kc = 0; kc < 5; ++kc) {
      int k  = kc * 4 + 2 * half;
      int i0 = k - m;
      int i1 = k + 1 - m;
      bx0[ky][kc] = (i0 >= 0 && i0 < 5) ? wl[ky * 5 + i0] : 0.0f;
      bx1[ky][kc] = (i1 >= 0 && i1 < 5) ? wl[ky * 5 + i1] : 0.0f;
    }
  }

  const int ty = (wave >> 2) * 16;  // 0 or 16 (row offset within LDS tile)
  const int tx = (wave & 3) * 16;   // 0,16,32,48 (col offset within LDS tile)

  v8f acc = {};
#pragma unroll
  for (int ky = 0; ky < 5; ++ky) {
#pragma unroll
    for (int kc = 0; kc < 5; ++kc) {
      int k = kc * 4 + 2 * half;
      v2f a;
      a.x = tile[ty + ky + m][tx + k];
      a.y = tile[ty + ky + m][tx + k + 1];
      v2f b;
      b.x = bx0[ky][kc];
      b.y = bx1[ky][kc];
      // D = A(16x4) * B(4x16) + C, fp32
      acc = __builtin_amdgcn_wmma_f32_16x16x4_f32(
          false, a, false, b, (short)0, acc, false, false);
    }
  }

  const float bv = wl[25];
  const int ox = bx + tx + m;
#pragma unroll
  for (int r = 0; r < 8; ++r) {
    int oy = by + ty + r + 8 * half;  // C layout: VGPR r holds M = r + 8*half
    if (oy < OUT_H && ox < OUT_W) {
      out[oy * OUT_W + ox] = acc[r] + bv;
    }
  }
}

extern "C" void kernel_launch(void* const* d_in, const int* in_sizes, int n_in,
                              void* d_out, int out_size, void* d_ws, size_t ws_size,
                              hipStream_t stream) {
  (void)in_sizes; (void)n_in; (void)out_size; (void)d_ws; (void)ws_size;
  const float* x    = (const float*)d_in[0];
  const float* wgt  = (const float*)d_in[1];
  const float* bias = (const float*)d_in[2];
  float* out        = (float*)d_out;

  dim3 grid((OUT_W + 63) / 64, (OUT_H + 31) / 32);
  dim3 block(256);
  conv5x5_wmma_f32<<<grid, block, 0, stream>>>(x, wgt, bias, out);
}